// DeepLinearTimeSeries_7584912245163
// MI455X (gfx1250) — compile-verified
//
#include <hip/hip_runtime.h>

typedef float v2f __attribute__((ext_vector_type(2)));
typedef float v4f __attribute__((ext_vector_type(4)));
typedef float v8f __attribute__((ext_vector_type(8)));

#define RWF 0.1f
#define HDIM 64
#define WSTRIDE 68          // padded row stride (floats) -> bank-conflict-free, 16B aligned (272B)
#define ROWS_PER_BLOCK 128
#define THREADS 128
#define NLAYER 399

__device__ __forceinline__ const float* layer_w(int l, const float* We,
                                                const float* Wt, const float* Wd) {
  if (l < 200) return We + (size_t)l * (HDIM * HDIM);
  if (l < 300) return Wt + (size_t)(l - 200) * (HDIM * HDIM);
  return Wd + (size_t)(l - 300) * (HDIM * HDIM);
}

__global__ __launch_bounds__(THREADS) void deep_linear_ts_kernel(
    const float* __restrict__ x, const float* __restrict__ We,
    const float* __restrict__ Wt, const float* __restrict__ Wd,
    const float* __restrict__ Wo, float* __restrict__ out) {
  extern __shared__ float smem[];
  float* wbuf = smem;                       // [2][64 * WSTRIDE]   weights, double-buffered
  float* abuf = smem + 2 * HDIM * WSTRIDE;  // [2][128 * WSTRIDE]  activations, ping-pong

  const int tid  = threadIdx.x;
  const int lane = tid & 31;
  const int wave = tid >> 5;
  const int kh   = lane >> 4;   // half-wave select (0/1)
  const int l16  = lane & 15;
  const size_t rowBase = (size_t)blockIdx.x * ROWS_PER_BLOCK;

  // ---- stage input tile: thread t owns token row t (row-major, padded) ----
  {
    const float* src = x + (rowBase + (size_t)tid) * HDIM;
    float* dst = abuf + tid * WSTRIDE;  // ping buffer 0 (layer 0 reads buf[0])
#pragma unroll
    for (int i = 0; i < HDIM / 4; ++i)
      *(v4f*)(dst + 4 * i) = *(const v4f*)(src + 4 * i);
  }

  // ---- prefetch layer-0 weights into registers: 8 x float4 per thread ----
  v4f wreg[8];
  {
    const float* wp = layer_w(0, We, Wt, Wd);
#pragma unroll
    for (int i = 0; i < 8; ++i)
      wreg[i] = *(const v4f*)(wp + (size_t)(tid + i * THREADS) * 4);
  }

  const int R = wave * 32;  // this wave's row base within the block tile

  for (int l = 0; l < NLAYER; ++l) {
    __syncthreads();  // everyone done reading wbuf[l&1] (last used at layer l-2)

    // residual handling per reference schedule:
    //   l==0,200   : no residual
    //   l==1,201   : residual = block input (two-layers-back state)
    //   otherwise  : residual = own input  -> fold rw into W diagonal
    const bool fold  = !(l == 0 || l == 1 || l == 200 || l == 201);
    const bool mode1 = (l == 1) || (l == 201);

    // ---- stage W into LDS as raw row-major [n][k] (B reads W[n][k]) ----
    {
      float* wdst = wbuf + (l & 1) * HDIM * WSTRIDE;
#pragma unroll
      for (int i = 0; i < 8; ++i) {
        const int idx = tid + i * THREADS;  // float4 index 0..1023
        const int n = idx >> 4;             // out-feature row
        const int c = idx & 15;             // k-block (covers k = 4c..4c+3)
        v4f v = wreg[i];
        if (fold && ((n >> 2) == c)) v[n & 3] += RWF;  // Wᵀ + rw·I
        *(v4f*)(wdst + n * WSTRIDE + 4 * c) = v;       // 272n+16c bytes: 16B aligned
      }
    }
    // ---- prefetch next layer's W (overlaps with this layer's WMMAs) ----
    if (l + 1 < NLAYER) {
      const float* wp = layer_w(l + 1, We, Wt, Wd);
#pragma unroll
      for (int i = 0; i < 8; ++i)
        wreg[i] = *(const v4f*)(wp + (size_t)(tid + i * THREADS) * 4);
    }
    __syncthreads();  // staged W visible to all waves

    const float* aIn  = abuf + (size_t)(l & 1) * ROWS_PER_BLOCK * WSTRIDE;
    float*       dOut = abuf + (size_t)((l + 1) & 1) * ROWS_PER_BLOCK * WSTRIDE;
    const float* wsrc = wbuf + (l & 1) * HDIM * WSTRIDE;

    v8f acc[2][4];
    if (mode1) {
      // two-layers-back activation lives in the (about to be overwritten) write buffer
#pragma unroll
      for (int m = 0; m < 2; ++m)
#pragma unroll
        for (int n = 0; n < 4; ++n)
#pragma unroll
          for (int r = 0; r < 8; ++r)
            acc[m][n][r] =
                RWF * dOut[(R + 16 * m + r + 8 * kh) * WSTRIDE + 16 * n + l16];
    } else {
#pragma unroll
      for (int m = 0; m < 2; ++m)
#pragma unroll
        for (int n = 0; n < 4; ++n)
          acc[m][n] = v8f{};
    }

    // ---- 32x64 += 32x64 * 64x64 via v_wmma_f32_16x16x4_f32 ----
#pragma unroll
    for (int k0 = 0; k0 < HDIM; k0 += 4) {
      // A frag (16x4): lanes 0-15 M=l16 K=k0..k0+1 ; lanes 16-31 M=l16 K=k0+2..k0+3
      v2f a0 = *(const v2f*)(aIn + (R + l16) * WSTRIDE + k0 + 2 * kh);
      v2f a1 = *(const v2f*)(aIn + (R + 16 + l16) * WSTRIDE + k0 + 2 * kh);
#pragma unroll
      for (int n = 0; n < 4; ++n) {
        // B frag (4x16): B[k][n'] = W[n'][k]; lanes 0-15 N=l16 K=k0..k0+1, upper half K=k0+2..k0+3
        v2f b = *(const v2f*)(wsrc + (16 * n + l16) * WSTRIDE + k0 + 2 * kh);
        acc[0][n] = __builtin_amdgcn_wmma_f32_16x16x4_f32(
            false, a0, false, b, (short)0, acc[0][n], false, false);
        acc[1][n] = __builtin_amdgcn_wmma_f32_16x16x4_f32(
            false, a1, false, b, (short)0, acc[1][n], false, false);
      }
    }

    // ---- write D (C-layout) back to the ping-pong activation buffer ----
#pragma unroll
    for (int m = 0; m < 2; ++m)
#pragma unroll
      for (int n = 0; n < 4; ++n)
#pragma unroll
        for (int r = 0; r < 8; ++r)
          dOut[(R + 16 * m + r + 8 * kh) * WSTRIDE + 16 * n + l16] = acc[m][n][r];
  }

  __syncthreads();
  // ---- final projection: out[row] = dot(act[row], W_out) ----
  const float* fin = abuf + (size_t)(NLAYER & 1) * ROWS_PER_BLOCK * WSTRIDE;
  float s = 0.f;
#pragma unroll
  for (int c = 0; c < HDIM; ++c)
    s = __builtin_fmaf(fin[tid * WSTRIDE + c], Wo[c], s);
  out[rowBase + tid] = s;
}

extern "C" void kernel_launch(void* const* d_in, const int* in_sizes, int n_in,
                              void* d_out, int out_size, void* d_ws, size_t ws_size,
                              hipStream_t stream) {
  (void)n_in; (void)d_ws; (void)ws_size; (void)out_size;
  const float* x  = (const float*)d_in[0];
  const float* We = (const float*)d_in[1];
  const float* Wt = (const float*)d_in[2];
  const float* Wd = (const float*)d_in[3];
  const float* Wo = (const float*)d_in[4];
  float* out = (float*)d_out;

  const int nrows = in_sizes[0] / HDIM;          // 128*2048 = 262144
  const int grid  = nrows / ROWS_PER_BLOCK;      // 2048 blocks
  const size_t shmem =
      (size_t)(2 * HDIM * WSTRIDE + 2 * ROWS_PER_BLOCK * WSTRIDE) * sizeof(float);  // 104,448 B

  // allow >64KB dynamic LDS (WGP has 320KB); idempotent + deterministic
  (void)hipFuncSetAttribute((const void*)deep_linear_ts_kernel,
                            hipFuncAttributeMaxDynamicSharedMemorySize, (int)shmem);

  deep_linear_ts_kernel<<<grid, THREADS, shmem, stream>>>(x, We, Wt, Wd, Wo, out);
}